// PointerDecoder_58256936403626
// MI455X (gfx1250) — compile-verified
//
#include <hip/hip_runtime.h>
#include <math.h>

#define B 32
#define N 512
#define H 256
#define T (N + 1)   // 513
#define G (4 * H)   // 1024

typedef __attribute__((ext_vector_type(2))) float v2f;
typedef __attribute__((ext_vector_type(8))) float v8f;
typedef __attribute__((ext_vector_type(4))) unsigned int v4u;
typedef __attribute__((ext_vector_type(8))) int v8i;
typedef __attribute__((ext_vector_type(4))) int v4i;

// CDNA5 transcendental tanh (VOP1 TRANS op). ISA: TRANS result needs 1
// independent op before use / source overwrite -> pad with v_nops inside asm.
__device__ __forceinline__ float fast_tanh(float x) {
  float r;
  asm volatile("v_tanh_f32 %0, %1\n\tv_nop\n\tv_nop" : "=v"(r) : "v"(x));
  return r;
}

__device__ __forceinline__ float sigmoidf(float x) {
  return 1.0f / (1.0f + __expf(-x));
}

// ---- TDM: flat 1D DMA of n_elems fp32 from global to LDS -----------------
// D# per CDNA5 ISA: group0 = {count=1 | lds_addr | global_addr |
// type=2}; group1 = {data_size=4B, tensor_dim0=tile_dim0=n, tensor_dim1=1,
// dim0_stride=n}; groups 2/3 zero (<=2D tensor).
// This toolchain declares the 6-arg builtin (clang-23 / therock-10 form):
//   (uint32x4 g0, int32x8 g1, int32x4 g2, int32x4 g3, int32x8 pad, i32 cpol)
__device__ __forceinline__ void tdm_load_1d_f32(unsigned lds_byte_off,
                                                const float* gptr,
                                                unsigned n) {
  unsigned long long ga = (unsigned long long)(size_t)gptr;
  v4u g0 = { 1u,                                              // count=1
             lds_byte_off,                                    // lds_addr [63:32]
             (unsigned)(ga & 0xFFFFFFFFu),                    // global_addr lo
             (unsigned)((ga >> 32) & 0x01FFFFFFu) | (2u << 30) }; // addr hi | type=2
  v8i g1 = { (int)(2u << 16),                                 // data_size=2 (4B)
             (int)((n & 0xFFFFu) << 16),                      // tensor_dim0[15:0] @bits63:48
             (int)((n >> 16) & 0xFFFFu) | (1 << 16),          // dim0 hi | tensor_dim1=1
             (int)((n & 0xFFFFu) << 16),                      // tile_dim0 @bits127:112
             0,                                               // tile_dim1/2 unused
             (int)n,                                          // tensor_dim0_stride lo32
             0, 0 };
  v4i z4 = { 0, 0, 0, 0 };
  v8i z8 = { 0, 0, 0, 0, 0, 0, 0, 0 };
  __builtin_amdgcn_tensor_load_to_lds(g0, g1, z4, z4, z8, 0);
}

// -------- init: h=c=0, mask=0, x = start_token broadcast ------------------
__global__ void k_init(const float* __restrict__ start, float* __restrict__ x,
                       float* __restrict__ h, float* __restrict__ c,
                       float* __restrict__ mask) {
  int i = blockIdx.x * blockDim.x + threadIdx.x;
  if (i < B * H) { h[i] = 0.f; c[i] = 0.f; x[i] = start[i % H]; }
  if (i < B * T) mask[i] = 0.f;
}

// -------- enc_proj[b,t,k] = sum_h ext[b,t,h] * W_a[k,h] -------------------
// GEMM M=B*T=16416, N=H, K=H via V_WMMA_F32_16X16X4_F32, one 16x16 tile/wave.
__global__ void k_enc_proj(const float* __restrict__ enc, const float* __restrict__ end_tok,
                           const float* __restrict__ Wa, float* __restrict__ enc_proj) {
  const int lane = threadIdx.x & 31;
  const int wave = threadIdx.x >> 5;
  const int NT = H / 16;                                  // 16 col-tiles
  const int tile = blockIdx.x * (blockDim.x >> 5) + wave; // exact grid, no tail
  const int tm = tile / NT, tn = tile % NT;
  const int half = lane >> 4, l = lane & 15;
  const int row = tm * 16 + l;                            // row in [0, B*T)
  const int bb = row / T, tt = row % T;
  const float* arow = (tt < N) ? (enc + ((size_t)bb * N + tt) * H) : end_tok;
  const int n = tn * 16 + l;                              // output column

  v8f acc = {};
  for (int kk = 0; kk < H; kk += 4) {
    const int k0 = kk + half * 2;                         // lanes 0-15: K=kk,kk+1; 16-31: kk+2,kk+3
    v2f a, b;
    a.x = arow[k0];          a.y = arow[k0 + 1];
    b.x = Wa[n * H + k0];    b.y = Wa[n * H + k0 + 1];    // B[k][n] = W_a[n][k]
    acc = __builtin_amdgcn_wmma_f32_16x16x4_f32(false, a, false, b, (short)0, acc,
                                                false, false);
  }
  const int m0 = tm * 16 + half * 8;                      // C layout: M = v (+8 upper lanes)
  #pragma unroll
  for (int v = 0; v < 8; ++v)
    enc_proj[(size_t)(m0 + v) * H + tn * 16 + l] = acc[v];
}

// -------- gates = [x|h] * [W_ih;W_hh]^T : M=32, N=1024, K=512 -------------
// A-tile (16 rows x 512 K = 32 KB) is shared by all 4 waves of the block
// (4 consecutive tn share tm): stage it once via TDM DMA into LDS, then
// read A-fragments from LDS while streaming the B (weight) tiles from L2.
__global__ void k_gates(const float* __restrict__ x, const float* __restrict__ h,
                        const float* __restrict__ Wih, const float* __restrict__ Whh,
                        float* __restrict__ gates) {
  __shared__ float aX[16 * H];                            // 16 KB
  __shared__ float aH[16 * H];                            // 16 KB
  const int lane = threadIdx.x & 31, wave = threadIdx.x >> 5;
  const int NT = G / 16;                                  // 64
  const int tile = blockIdx.x * 4 + wave;                 // 128 tiles, exact
  const int tm = tile / NT, tn = tile % NT;               // tm uniform in block
  const int half = lane >> 4, l = lane & 15;
  const int n = tn * 16 + l;                              // gate column

  if (threadIdx.x == 0) {
    tdm_load_1d_f32((unsigned)(size_t)aX, x + tm * 16 * H, 16 * H);
    tdm_load_1d_f32((unsigned)(size_t)aH, h + tm * 16 * H, 16 * H);
    __builtin_amdgcn_s_wait_tensorcnt(0);
  }
  __syncthreads();                                        // LDS tile visible

  v8f acc = {};
  for (int kk = 0; kk < 2 * H; kk += 4) {
    const int k0 = kk + half * 2;                         // k0,k0+1 never straddle H
    v2f a, b;
    const float* as = (k0 < H) ? (aX + l * H + k0) : (aH + l * H + (k0 - H));
    const float* bs = (k0 < H) ? (Wih + n * H + k0) : (Whh + n * H + (k0 - H));
    a.x = as[0]; a.y = as[1];
    b.x = bs[0]; b.y = bs[1];
    acc = __builtin_amdgcn_wmma_f32_16x16x4_f32(false, a, false, b, (short)0, acc,
                                                false, false);
  }
  const int m0 = tm * 16 + half * 8;
  #pragma unroll
  for (int v = 0; v < 8; ++v)
    gates[(m0 + v) * G + tn * 16 + l] = acc[v];
}

// -------- LSTM pointwise (torch gate order i,f,g,o) -----------------------
__global__ void k_lstm(const float* __restrict__ gates, const float* __restrict__ b_ih,
                       const float* __restrict__ b_hh, float* __restrict__ h,
                       float* __restrict__ c) {
  int i = blockIdx.x * blockDim.x + threadIdx.x;
  if (i >= B * H) return;
  int b = i / H, k = i % H;
  const float* gr = gates + b * G;
  float gi = gr[k]           + b_ih[k]           + b_hh[k];
  float gf = gr[k + H]       + b_ih[k + H]       + b_hh[k + H];
  float gg = gr[k + 2 * H]   + b_ih[k + 2 * H]   + b_hh[k + 2 * H];
  float go = gr[k + 3 * H]   + b_ih[k + 3 * H]   + b_hh[k + 3 * H];
  float c2 = sigmoidf(gf) * c[i] + sigmoidf(gi) * fast_tanh(gg);
  c[i] = c2;
  h[i] = sigmoidf(go) * fast_tanh(c2);
}

// -------- q = h * U_a^T : M=32, N=256, K=256 ------------------------------
__global__ void k_q(const float* __restrict__ h, const float* __restrict__ Ua,
                    float* __restrict__ q) {
  const int lane = threadIdx.x & 31, wave = threadIdx.x >> 5;
  const int NT = H / 16;                                  // 16
  const int tile = blockIdx.x * (blockDim.x >> 5) + wave; // 32 tiles, exact
  const int tm = tile / NT, tn = tile % NT;
  const int half = lane >> 4, l = lane & 15;
  const int m = tm * 16 + l, n = tn * 16 + l;
  v8f acc = {};
  for (int kk = 0; kk < H; kk += 4) {
    const int k0 = kk + half * 2;
    v2f a, b;
    a.x = h[m * H + k0];   a.y = h[m * H + k0 + 1];
    b.x = Ua[n * H + k0];  b.y = Ua[n * H + k0 + 1];
    acc = __builtin_amdgcn_wmma_f32_16x16x4_f32(false, a, false, b, (short)0, acc,
                                                false, false);
  }
  const int m0 = tm * 16 + half * 8;
  #pragma unroll
  for (int v = 0; v < 8; ++v)
    q[(m0 + v) * H + tn * 16 + l] = acc[v];
}

// -------- scores[b,t] = v_a . tanh(enc_proj[b,t,:] + q[b,:]) --------------
// One wave per (b,t) row; each lane owns 8 contiguous floats -> b128 loads
// (this is the L2-bandwidth-bound phase: 16.8 MB of enc_proj per step).
__global__ void k_scores(const float* __restrict__ enc_proj, const float* __restrict__ q,
                         const float* __restrict__ va, float* __restrict__ scores) {
  const int lane = threadIdx.x & 31, wave = threadIdx.x >> 5;
  const int r = blockIdx.x * (blockDim.x >> 5) + wave;    // B*T rows, exact grid
  const int b = r / T;
  const float4* ep4 = (const float4*)(enc_proj + (size_t)r * H) + lane * 2;
  const float4* q4  = (const float4*)(q + b * H) + lane * 2;
  const float4* va4 = (const float4*)va + lane * 2;
  float s = 0.f;
  #pragma unroll
  for (int j = 0; j < 2; ++j) {                           // 2 x float4 per lane
    float4 e = ep4[j], qq = q4[j], vv = va4[j];
    s += vv.x * fast_tanh(e.x + qq.x);
    s += vv.y * fast_tanh(e.y + qq.y);
    s += vv.z * fast_tanh(e.z + qq.z);
    s += vv.w * fast_tanh(e.w + qq.w);
  }
  #pragma unroll
  for (int off = 16; off; off >>= 1) s += __shfl_xor(s, off, 32);
  if (lane == 0) scores[r] = s;
}

// -------- per-batch: mask, log_softmax -> out, argmax, next input ---------
__global__ void k_select(const float* __restrict__ scores, float* __restrict__ mask,
                         const float* __restrict__ enc, const float* __restrict__ end_tok,
                         float* __restrict__ x, float* __restrict__ out_lp,
                         float* __restrict__ out_sel, int step) {
  const int b = blockIdx.x;
  const int tid = threadIdx.x;                            // 256 threads
  __shared__ float red[256];
  __shared__ int   redi[256];
  __shared__ float s_max, s_logsum;
  __shared__ int   s_idx;

  const float* sr = scores + b * T;
  const float* mr = mask + b * T;
  const float NINF = -__builtin_inff();

  // masked max + first-index argmax
  float best = NINF; int bi = T;
  for (int t = tid; t < T; t += 256) {
    float v = (mr[t] == 1.0f) ? NINF : sr[t];
    if (v > best || (v == best && t < bi)) { best = v; bi = t; }
  }
  red[tid] = best; redi[tid] = bi;
  __syncthreads();
  for (int off = 128; off; off >>= 1) {
    if (tid < off) {
      float ov = red[tid + off]; int oi = redi[tid + off];
      if (ov > red[tid] || (ov == red[tid] && oi < redi[tid])) {
        red[tid] = ov; redi[tid] = oi;
      }
    }
    __syncthreads();
  }
  if (tid == 0) { s_max = red[0]; s_idx = redi[0]; }
  __syncthreads();
  const float mx = s_max;

  // sum of exp
  float se = 0.f;
  for (int t = tid; t < T; t += 256) {
    float v = (mr[t] == 1.0f) ? NINF : sr[t];
    se += __expf(v - mx);
  }
  red[tid] = se;
  __syncthreads();
  for (int off = 128; off; off >>= 1) {
    if (tid < off) red[tid] += red[tid + off];
    __syncthreads();
  }
  if (tid == 0) s_logsum = __logf(red[0]);
  __syncthreads();
  const float ls = s_logsum;

  float* lp = out_lp + ((size_t)b * T + step) * T;
  for (int t = tid; t < T; t += 256) {
    float v = (mr[t] == 1.0f) ? NINF : sr[t];
    lp[t] = v - mx - ls;
  }
  __syncthreads();                                        // all mask reads done

  const int idx = s_idx;
  const bool sel = idx < N;                               // end token never masked
  if (tid == 0) {
    out_sel[b * T + step] = (float)idx;
    if (sel) mask[b * T + idx] = 1.0f;
  }
  const int gi = sel ? idx : (N - 1);
  const float* src = sel ? (enc + ((size_t)b * N + gi) * H) : end_tok;
  for (int k = tid; k < H; k += 256) x[b * H + k] = src[k];
}

extern "C" void kernel_launch(void* const* d_in, const int* in_sizes, int n_in,
                              void* d_out, int out_size, void* d_ws, size_t ws_size,
                              hipStream_t stream) {
  (void)in_sizes; (void)n_in; (void)out_size; (void)ws_size;
  const float* enc = (const float*)d_in[0];
  const float* Wih = (const float*)d_in[1];
  const float* Whh = (const float*)d_in[2];
  const float* bih = (const float*)d_in[3];
  const float* bhh = (const float*)d_in[4];
  const float* Wa  = (const float*)d_in[5];
  const float* Ua  = (const float*)d_in[6];
  const float* va  = (const float*)d_in[7];
  const float* st  = (const float*)d_in[8];
  const float* et  = (const float*)d_in[9];

  float* ws = (float*)d_ws;
  float* enc_proj = ws; ws += (size_t)B * T * H;   // 16.8 MB, L2-resident
  float* x        = ws; ws += B * H;
  float* h        = ws; ws += B * H;
  float* c        = ws; ws += B * H;
  float* gates    = ws; ws += B * G;
  float* q        = ws; ws += B * H;
  float* scores   = ws; ws += B * T;
  float* mask     = ws; ws += B * T;

  float* out_lp  = (float*)d_out;                          // [B, T, T]
  float* out_sel = (float*)d_out + (size_t)B * T * T;      // [B, T]

  k_init<<<(B * T + 255) / 256, 256, 0, stream>>>(st, x, h, c, mask);

  {
    int tiles = (B * T / 16) * (H / 16);                   // 16416, /4 exact
    k_enc_proj<<<tiles / 4, 128, 0, stream>>>(enc, et, Wa, enc_proj);
  }

  for (int step = 0; step < T; ++step) {
    k_gates<<<(2 * (G / 16)) / 4, 128, 0, stream>>>(x, h, Wih, Whh, gates);
    k_lstm<<<(B * H + 255) / 256, 256, 0, stream>>>(gates, bih, bhh, h, c);
    k_q<<<(2 * (H / 16)) / 4, 128, 0, stream>>>(h, Ua, q);
    k_scores<<<(B * T) / 4, 128, 0, stream>>>(enc_proj, q, va, scores);
    k_select<<<B, 256, 0, stream>>>(scores, mask, enc, et, x, out_lp, out_sel, step);
  }
}